// HighOrderNet_37752762531918
// MI455X (gfx1250) — compile-verified
//
#include <hip/hip_runtime.h>

// HighOrderNet: out[n,o] = sum_h (prod_{i!=msg_to} inp[i,n,h]) * params[id[n],h,o] + bias[id[n],o]
// id[n] = x[fact[n,0],1]*13 + x[fact[n,0],2];  N=1024, H=O=512, 169 weight matrices.
//
// Kernel 1: counting-sort rows by weight id, build (group, mtile) worklist.
// Kernel 2: grouped f32 GEMM with V_WMMA_F32_16X16X4_F32. Per block: 16 rows of
// one id staged as fact_prod into LDS once (generic ORD/msg_to handled there),
// 4 waves x 4 col-tiles = 256 output columns. Hot loop: ds_load_b64 (A) +
// 8x global_load_b32 (B, L2-resident) + 4 WMMA per K-step of 4.

#define NPAIR 169  // 13*13 weight matrices

typedef __attribute__((ext_vector_type(2))) float v2f;
typedef __attribute__((ext_vector_type(8))) float v8f;

// ---------------------------------------------------------------------------
// Kernel 1: ids, histogram, prefix sum, permutation, (group, mtile) worklist.
// ---------------------------------------------------------------------------
__global__ void __launch_bounds__(256)
hon_prep(const int* __restrict__ x, const int* __restrict__ fact,
         int* __restrict__ ids, int* __restrict__ perm,
         int* __restrict__ gstart, int* __restrict__ work,
         int N, int wcap)
{
    __shared__ int cnt[NPAIR];
    __shared__ int pos[NPAIR];
    const int tid = threadIdx.x;

    for (int g = tid; g < NPAIR; g += blockDim.x) cnt[g] = 0;
    __syncthreads();

    for (int n = tid; n < N; n += blockDim.x) {
        const int a  = fact[2 * n];                    // fact[n,0]
        const int id = x[3 * a + 1] * 13 + x[3 * a + 2];
        ids[n] = id;
        atomicAdd(&cnt[id], 1);
    }
    __syncthreads();

    if (tid == 0) {
        int acc = 0, w = 0;
        for (int g = 0; g < NPAIR; ++g) {
            gstart[g] = acc;
            pos[g]    = acc;
            const int c  = cnt[g];
            const int nt = (c + 15) >> 4;              // 16-row m-tiles
            for (int t = 0; t < nt && w < wcap; ++t, ++w) {
                work[2 * w]     = g;
                work[2 * w + 1] = t;
            }
            acc += c;
        }
        gstart[NPAIR] = acc;
        for (; w < wcap; ++w) { work[2 * w] = -1; work[2 * w + 1] = 0; }
    }
    __syncthreads();

    for (int n = tid; n < N; n += blockDim.x) {
        const int p = atomicAdd(&pos[ids[n]], 1);
        perm[p] = n;
    }
}

// ---------------------------------------------------------------------------
// Kernel 2: grouped GEMM, 128 threads = 4 waves per block.
// A (16x4 f32): lane L -> row M=L&15, K pair {(L>>4)*2, (L>>4)*2+1}.
// B (4x16 f32): lane L -> col N=L&15, same K pair.
// D (16x16 f32): VGPR r -> row r (lanes 0-15) / r+8 (lanes 16-31), col L&15.
// Block covers 256 output columns: wave w -> cols [by*256 + w*64, +64).
// ---------------------------------------------------------------------------
__global__ void __launch_bounds__(128)
hon_gemm(const float* __restrict__ inp, const float* __restrict__ params,
         const float* __restrict__ bias, const int* __restrict__ perm,
         const int* __restrict__ gstart, const int* __restrict__ work,
         const int* __restrict__ msg_to_p, float* __restrict__ out,
         int N, int H, int O, int ORD)
{
    extern __shared__ float As[];                       // [16][H] fact_prod tile

    const int g = work[2 * blockIdx.x];
    if (g < 0) return;                                  // padded worklist entry
    const int mt     = work[2 * blockIdx.x + 1];
    const int rowbeg = gstart[g] + mt * 16;
    int cnt = gstart[g + 1] - rowbeg;
    cnt = cnt > 16 ? 16 : cnt;
    const int msg_to = msg_to_p[0];

    // ---- Stage fact_prod[16 rows][H] into LDS (float4 vectorized). --------
    {
        const int HQ = H >> 2;                          // float4s per row
        for (int idx = threadIdx.x; idx < 16 * HQ; idx += blockDim.x) {
            const int r  = idx / HQ;
            const int kq = idx - r * HQ;
            float4 p = make_float4(0.f, 0.f, 0.f, 0.f); // zero pad rows
            if (r < cnt) {
                const int n = perm[rowbeg + r];
                p = make_float4(1.f, 1.f, 1.f, 1.f);
                for (int i = 0; i < ORD; ++i) {         // generic keep-set
                    if (i == msg_to) continue;
                    const float4 v =
                        ((const float4*)(inp + ((size_t)i * N + n) * H))[kq];
                    p.x *= v.x; p.y *= v.y; p.z *= v.z; p.w *= v.w;
                }
            }
            ((float4*)(As + r * H))[kq] = p;
        }
    }
    __syncthreads();

    // ---- Grouped GEMM over K with f32 WMMA. --------------------------------
    const int lane   = threadIdx.x & 31;
    const int waveid = threadIdx.x >> 5;
    const int m      = lane & 15;
    const int khalf  = (lane >> 4) << 1;                // 0 or 2

    const float* W       = params + (size_t)g * H * O;
    const int    colbase = blockIdx.y * 256 + waveid * 64 + m;
    const float* aptr    = As + m * H + khalf;          // 8B aligned (khalf even)

    v8f acc[4];
#pragma unroll
    for (int t = 0; t < 4; ++t) acc[t] = (v8f){0, 0, 0, 0, 0, 0, 0, 0};

#pragma unroll 2
    for (int kb = 0; kb < H; kb += 4) {
        const int k0 = kb + khalf;
        const v2f a  = *(const v2f*)(aptr + kb);        // ds_load_b64

        const float* Wk0 = W + (size_t)k0 * O + colbase;
        const float* Wk1 = Wk0 + O;
#pragma unroll
        for (int t = 0; t < 4; ++t) {                   // 4 col tiles reuse A
            v2f b; b.x = Wk0[t * 16]; b.y = Wk1[t * 16];
            acc[t] = __builtin_amdgcn_wmma_f32_16x16x4_f32(
                false, a, false, b, (short)0, acc[t], false, false);
        }
    }

    // ---- Bias + store. -----------------------------------------------------
    const int mbase = (lane >> 4) << 3;                 // D rows of this half
    int nrow[8];
#pragma unroll
    for (int r = 0; r < 8; ++r) {
        const int mr = mbase + r;
        nrow[r] = (mr < cnt) ? perm[rowbeg + mr] : -1;
    }

#pragma unroll
    for (int t = 0; t < 4; ++t) {
        const int   col = colbase + t * 16;
        const float bv  = bias[(size_t)g * O + col];
#pragma unroll
        for (int r = 0; r < 8; ++r) {
            if (nrow[r] >= 0)
                out[(size_t)nrow[r] * O + col] = acc[t][r] + bv;
        }
    }
}

// ---------------------------------------------------------------------------
extern "C" void kernel_launch(void* const* d_in, const int* in_sizes, int n_in,
                              void* d_out, int out_size, void* d_ws, size_t ws_size,
                              hipStream_t stream)
{
    const int*   x      = (const int*)  d_in[0];   // [M,3] int32
    const int*   fact   = (const int*)  d_in[1];   // [N,2] int32
    const float* inp    = (const float*)d_in[2];   // [ORD,N,H] f32
    const float* params = (const float*)d_in[3];   // [169,H,O] f32
    const float* bias   = (const float*)d_in[4];   // [169,1,O] f32
    const int*   msg_to = (const int*)  d_in[5];   // scalar on device
    float*       out    = (float*)d_out;           // [N,O] f32

    const int N   = in_sizes[1] / 2;
    const int O   = in_sizes[4] / NPAIR;
    const int H   = in_sizes[3] / (NPAIR * O);
    const int ORD = in_sizes[2] / (N * H);

    const int wcap = NPAIR + (N + 15) / 16;        // >= total m-tiles

    int* ws     = (int*)d_ws;                      // ~11 KB of scratch
    int* ids    = ws;                              // [N]
    int* perm   = ws + N;                          // [N]
    int* gstart = ws + 2 * N;                      // [NPAIR+1]
    int* work   = ws + 2 * N + NPAIR + 1;          // [2*wcap]

    hipLaunchKernelGGL(hon_prep, dim3(1), dim3(256), 0, stream,
                       x, fact, ids, perm, gstart, work, N, wcap);

    const size_t lds_bytes = (size_t)16 * H * sizeof(float);
    dim3 grid(wcap, O / 256);                      // O=512 -> grid.y=2
    hipLaunchKernelGGL(hon_gemm, grid, dim3(128), lds_bytes, stream,
                       inp, params, bias, perm, gstart, work, msg_to, out,
                       N, H, O, ORD);
}